// HybridEstimatorQNN_65481071409645
// MI455X (gfx1250) — compile-verified
//
#include <hip/hip_runtime.h>
#include <hip/hip_bf16.h>

typedef float v2f __attribute__((ext_vector_type(2)));
typedef float v8f __attribute__((ext_vector_type(8)));

#if __has_builtin(__builtin_amdgcn_tanhf)
__device__ __forceinline__ float fast_tanh(float x) { return __builtin_amdgcn_tanhf(x); }
#else
__device__ __forceinline__ float fast_tanh(float x) { return tanhf(x); }
#endif

// Closed-form quanvolution feature for one 2x2 patch.
// m_q = (pix_q > 127) ? -cos(theta_q) : +cos(theta_q)
// feat = 0.5 - (m0m1 + m0m1m2 + m0m1m2m3 + m1m2m3)/8
__device__ __forceinline__ float quanv_feat(float p0, float p1, float p2, float p3,
                                            float c0, float c1, float c2, float c3) {
  const float TH = 127.0f;
  float m0 = (p0 > TH) ? -c0 : c0;
  float m1 = (p1 > TH) ? -c1 : c1;
  float m2 = (p2 > TH) ? -c2 : c2;
  float m3 = (p3 > TH) ? -c3 : c3;
  float m01 = m0 * m1;
  float m23 = m2 * m3;
  float t = fmaf(m01, 1.0f + m2 + m23, m1 * m23);
  return fmaf(-0.125f, t, 0.5f);
}

__global__ __launch_bounds__(256) void quanv_mlp_kernel(
    const float* __restrict__ images, const float* __restrict__ theta,
    const float* __restrict__ W1, const float* __restrict__ b1,
    const float* __restrict__ W2, const float* __restrict__ b2,
    const float* __restrict__ W3, const float* __restrict__ b3,
    float* __restrict__ out, int nImages) {
  const int lane   = threadIdx.x & 31;
  const int waveId = (int)((blockIdx.x * blockDim.x + threadIdx.x) >> 5);
  const int base   = waveId * 32;            // 32 images per wave (2 WMMA tiles)
  if (base >= nImages) return;

  const float c0 = __cosf(theta[0]);
  const float c1 = __cosf(theta[1]);
  const float c2 = __cosf(theta[2]);
  const float c3 = __cosf(theta[3]);

  const int m     = lane & 15;   // image-in-tile / A-matrix row
  const int khalf = lane >> 4;   // 0: patches {0,1} (rows 0-1), 1: patches {2,3} (rows 1-2)

  // ---- A operand = W1^T, 16x4 f32, rows 8..15 zero ----
  // A layout: lanes 0-15 hold K=0(v0),K=1(v1); lanes 16-31 hold K=2,K=3 for the same M=lane&15.
  v2f a;
  a.x = (m < 8) ? W1[(2 * khalf + 0) * 8 + m] : 0.0f;
  a.y = (m < 8) ? W1[(2 * khalf + 1) * 8 + m] : 0.0f;

  // ---- C preload = layer-1 bias (C layout: vgpr r -> M=r lanes 0-15, M=8+r lanes 16-31) ----
  v8f cc0, cc1;
  #pragma unroll
  for (int r = 0; r < 8; ++r) {
    float bv = (lane < 16) ? b1[r] : 0.0f;
    cc0[r] = bv;
    cc1[r] = bv;
  }

  // ---- B operands = feats^T (4x16): lane computes the 2 patch features it owns ----
  // Patches p = 2*khalf + j (j=0,1) use the 6 consecutive pixels [3*khalf .. 3*khalf+5].
  const int last = nImages - 1;
  int img0 = base + m;            if (img0 > last) img0 = last;
  int img1 = base + 16 + m;       if (img1 > last) img1 = last;

  v2f bt0, bt1;
  {
    const float* px = images + (size_t)img0 * 9 + 3 * khalf;
    float w0 = px[0], w1 = px[1], w2 = px[2], w3 = px[3], w4 = px[4], w5 = px[5];
    bt0.x = quanv_feat(w0, w1, w3, w4, c0, c1, c2, c3);
    bt0.y = quanv_feat(w1, w2, w4, w5, c0, c1, c2, c3);
  }
  {
    const float* px = images + (size_t)img1 * 9 + 3 * khalf;
    float w0 = px[0], w1 = px[1], w2 = px[2], w3 = px[3], w4 = px[4], w5 = px[5];
    bt1.x = quanv_feat(w0, w1, w3, w4, c0, c1, c2, c3);
    bt1.y = quanv_feat(w1, w2, w4, w5, c0, c1, c2, c3);
  }

  // ---- Layer 1 on the matrix pipe: h_pre = W1^T @ feats^T + b1 ----
  cc0 = __builtin_amdgcn_wmma_f32_16x16x4_f32(false, a, false, bt0, (short)0, cc0, false, false);
  cc1 = __builtin_amdgcn_wmma_f32_16x16x4_f32(false, a, false, bt1, (short)0, cc1, false, false);

  // Result: lane n (0-15) holds h_pre[0..7] of image (tileBase+n) in cc[0..7].
  // Move tile-1 hidden vectors into the upper half so every lane owns one image.
  float h[8];
  #pragma unroll
  for (int r = 0; r < 8; ++r) {
    float other = __shfl_xor(cc1[r], 16, 32);
    float pre   = (lane < 16) ? cc0[r] : other;
    h[r] = fast_tanh(pre);
  }

  // ---- Tail MLP (8 -> 4 -> 1), per-lane scalar (weights are uniform -> scalar loads) ----
  float g[4];
  #pragma unroll
  for (int j = 0; j < 4; ++j) {
    float s = b2[j];
    #pragma unroll
    for (int r = 0; r < 8; ++r) s = fmaf(h[r], W2[r * 4 + j], s);
    g[j] = fast_tanh(s);
  }
  float y = b3[0];
  #pragma unroll
  for (int j = 0; j < 4; ++j) y = fmaf(g[j], W3[j], y);

  const int img = base + lane;   // lanes 0-15 -> tile0, lanes 16-31 -> tile1
  if (img < nImages) out[img] = y;
}

extern "C" void kernel_launch(void* const* d_in, const int* in_sizes, int n_in,
                              void* d_out, int out_size, void* d_ws, size_t ws_size,
                              hipStream_t stream) {
  const float* images = (const float*)d_in[0];
  const float* theta  = (const float*)d_in[1];
  const float* W1     = (const float*)d_in[2];
  const float* b1     = (const float*)d_in[3];
  const float* W2     = (const float*)d_in[4];
  const float* b2     = (const float*)d_in[5];
  const float* W3     = (const float*)d_in[6];
  const float* b3     = (const float*)d_in[7];
  float* out = (float*)d_out;

  const int nImages = in_sizes[0] / 9;          // (B,1,3,3) float32
  const int waves   = (nImages + 31) / 32;      // 32 images per wave
  const long long threads = (long long)waves * 32;
  const int block = 256;
  const int grid  = (int)((threads + block - 1) / block);

  quanv_mlp_kernel<<<grid, block, 0, stream>>>(images, theta, W1, b1, W2, b2, W3, b3,
                                               out, nImages);
}